// ModalityConfidenceModule_46308337386009
// MI455X (gfx1250) — compile-verified
//
#include <hip/hip_runtime.h>
#include <cstdint>

// ---------------- constants ----------------
#define Wd    512
#define HWd   262144            // 512*512
#define Bd    32
#define RT1   8                 // rows per RGB tile
#define RP1   10                // staged rows (RT1 + 2 halo)
#define NT1   64                // RGB tiles per image (512/8)
#define RT2   16                // rows per depth tile
#define NT2   32                // depth tiles per image
#define EPSF  1e-4f

typedef __attribute__((ext_vector_type(2))) float    v2f;
typedef __attribute__((ext_vector_type(8))) float    v8f;
typedef __attribute__((ext_vector_type(4))) unsigned v4u;
typedef __attribute__((ext_vector_type(4))) int      v4i;
typedef __attribute__((ext_vector_type(8))) int      v8i;

// ---------------- TDM: tensor DMA global -> LDS (CDNA5) ----------------
// Builds a D# (4 SGPR groups) for a packed up-to-3D tile load.
// data_size = 4B. Strides in element units. lds destination is packed.
// Toolchain uses the 6-arg builtin: (v4u, v8i, v4i, v4i, v8i, i32 cpol).
__device__ __forceinline__ void tdm_load(unsigned lds_addr, const void* gptr,
                                         unsigned td0, unsigned td1, unsigned td2,
                                         unsigned tl0, unsigned tl1, unsigned tl2,
                                         unsigned long long s0, unsigned long long s1)
{
  unsigned long long ga = (unsigned long long)(uintptr_t)gptr;
  v4u g0;
  g0[0] = 1u;                                                   // count=1, no gather
  g0[1] = lds_addr;                                             // lds_addr[31:0]
  g0[2] = (unsigned)(ga & 0xFFFFFFFFull);                       // global_addr lo
  g0[3] = (unsigned)((ga >> 32) & 0x1FFFFFFull) | (2u << 30);   // global_addr hi | type=2
  v8i g1;
  g1[0] = (int)(2u << 16);                                      // wg_mask=0, data_size=4B
  g1[1] = (int)((td0 & 0xFFFFu) << 16);                         // tensor_dim0 lo16
  g1[2] = (int)((td0 >> 16) | ((td1 & 0xFFFFu) << 16));         // td0 hi16 | td1 lo16
  g1[3] = (int)((td1 >> 16) | (tl0 << 16));                     // td1 hi16 | tile_dim0
  g1[4] = (int)(tl1 | (tl2 << 16));                             // tile_dim1 | tile_dim2
  g1[5] = (int)(s0 & 0xFFFFFFFFull);                            // dim0_stride lo32
  g1[6] = (int)(((s0 >> 32) & 0xFFFFull) | ((s1 & 0xFFFFull) << 16)); // s0 hi16 | s1 lo16
  g1[7] = (int)((s1 >> 16) & 0xFFFFFFFFull);                    // dim1_stride hi32
  v4i g2; g2[0] = (int)td2; g2[1] = 0; g2[2] = 0; g2[3] = 0;    // tensor_dim2
  v4i g3; g3[0] = 0; g3[1] = 0; g3[2] = 0; g3[3] = 0;
  v8i g4; g4[0]=0; g4[1]=0; g4[2]=0; g4[3]=0; g4[4]=0; g4[5]=0; g4[6]=0; g4[7]=0;
  __builtin_amdgcn_tensor_load_to_lds(g0, g1, g2, g3, g4, 0);
}

// ---------------- wave32 shuffle reduction ----------------
__device__ __forceinline__ float wredf(float v) {
#pragma unroll
  for (int m = 16; m > 0; m >>= 1) v += __shfl_xor(v, m, 32);
  return v;
}

// ---------------- WMMA 32-lane sum (V_WMMA_F32_16X16X4_F32 tree) ----------------
__device__ __forceinline__ v8f wmma4(v2f a, v2f b, v8f c) {
  return __builtin_amdgcn_wmma_f32_16x16x4_f32(false, a, false, b, (short)0, c,
                                               false, false);
}
// A ones-B WMMA sums the 4 K-slots of each row. Chaining twice reduces all 32
// lane values; result is broadcast to every lane (no readlane / LDS needed).
__device__ __forceinline__ float waveSum32(float x) {
  v2f a;    a[0] = x;   a[1] = 0.f;
  v2f ones; ones[0] = 1.f; ones[1] = 1.f;
  v8f z = {};
  v8f d = wmma4(a, ones, z);          // d rows m: x[m] + x[m+16]
  v2f p01; p01[0] = d[0]; p01[1] = d[1];
  v2f p23; p23[0] = d[2]; p23[1] = d[3];
  v2f p45; p45[0] = d[4]; p45[1] = d[5];
  v2f p67; p67[0] = d[6]; p67[1] = d[7];
  v8f t0 = wmma4(p01, ones, z);       // r0+r1+r8+r9   (broadcast all lanes)
  v8f t1 = wmma4(p23, ones, z);       // r2+r3+r10+r11
  v8f t2 = wmma4(p45, ones, z);       // r4+r5+r12+r13
  v8f t3 = wmma4(p67, ones, z);       // r6+r7+r14+r15
  return t0[0] + t1[0] + t2[0] + t3[0];
}

// ---------------- kernel 0: zero histogram accumulators ----------------
__global__ void mcm_zero_hist(unsigned* __restrict__ h) {
  h[blockIdx.x * 256 + threadIdx.x] = 0u;
}

// ---------------- kernel 1: RGB gray/Laplacian/histogram reduce ----------------
__global__ __launch_bounds__(256) void mcm_rgb_reduce(
    const float* __restrict__ rgb, float* __restrict__ lapSum,
    float* __restrict__ lapSq, unsigned* __restrict__ ghist)
{
  __shared__ float    ch[3 * RP1 * Wd];   // staged channel tile (60 KB)
  __shared__ unsigned lh[256];            // block histogram
  __shared__ float    red[16];            // cross-wave partials

  const int tile = blockIdx.x, b = blockIdx.y, tid = threadIdx.x;
  const int r0   = tile * RT1;
  const int lo   = (tile == 0) ? 0 : (r0 - 1);
  const int hi   = (tile == NT1 - 1) ? Wd : (r0 + RT1 + 1);
  const int nr   = hi - lo;
  const int skip = (tile == 0) ? 1 : 0;

  lh[tid] = 0u;

  // wave 0 issues the TDM DMA (wave-level instruction; one issue per block)
  if (tid < 32) {
    const float* base = rgb + (size_t)b * 3 * HWd;
    unsigned ldsb = (unsigned)(uintptr_t)(void*)ch;
    if (nr == RP1) {
      // interior tile: single 3D tensor load (512 x 10 x 3), plane stride HW
      tdm_load(ldsb, base + (size_t)lo * Wd,
               Wd, RP1, 3, Wd, RP1, 3,
               (unsigned long long)Wd, (unsigned long long)HWd);
    } else {
      // boundary: one 2D load per channel into the shifted row window
      for (int c = 0; c < 3; ++c)
        tdm_load(ldsb + (unsigned)((c * RP1 + skip) * Wd * 4),
                 base + (size_t)c * HWd + (size_t)lo * Wd,
                 Wd, (unsigned)nr, 1, Wd, (unsigned)nr, 1,
                 (unsigned long long)Wd, 0ull);
    }
  }
  // zero halo row that has no source data (top/bottom image edge)
  if (nr != RP1) {
    int zr = (tile == 0) ? 0 : (RP1 - 1);
    for (int i = tid; i < 3 * Wd; i += 256)
      ch[((i >> 9) * RP1 + zr) * Wd + (i & (Wd - 1))] = 0.f;
  }
  if (tid < 32) __builtin_amdgcn_s_wait_tensorcnt(0);
  __syncthreads();

  // gray(x) = .299R + .587G + .114B ; lap(gray) from per-channel LDS taps
#define GRAY(LR, CC) (0.299f * ch[(0 * RP1 + (LR)) * Wd + (CC)] + \
                      0.587f * ch[(1 * RP1 + (LR)) * Wd + (CC)] + \
                      0.114f * ch[(2 * RP1 + (LR)) * Wd + (CC)])
  float s1 = 0.f, s2 = 0.f;
#pragma unroll 4
  for (int i = 0; i < 16; ++i) {
    int p  = tid + i * 256;
    int cr = p >> 9;            // 0..7
    int cc = p & (Wd - 1);
    int lr = cr + 1;            // LDS row (halo at 0 and RP1-1)
    float c0  = GRAY(lr, cc);
    float lap = GRAY(lr - 1, cc) + GRAY(lr + 1, cc) - 4.f * c0;
    if (cc > 0)      lap += GRAY(lr, cc - 1);
    if (cc < Wd - 1) lap += GRAY(lr, cc + 1);
    s1 += lap;
    s2 += lap * lap;
    float gv = fminf(fmaxf(c0 * 255.f, 0.f), 255.f);
    atomicAdd(&lh[(int)gv], 1u);
  }
#undef GRAY

  // block reduction of Laplacian sums -> per-block slot (deterministic)
  s1 = wredf(s1);
  s2 = wredf(s2);
  int wave = tid >> 5, lane = tid & 31;
  if (lane == 0) { red[wave] = s1; red[8 + wave] = s2; }
  __syncthreads();
  if (tid == 0) {
    float a = 0.f, c = 0.f;
    for (int w = 0; w < 8; ++w) { a += red[w]; c += red[8 + w]; }
    lapSum[b * NT1 + tile] = a;
    lapSq [b * NT1 + tile] = c;
  }
  // integer histogram flush (deterministic)
  atomicAdd(&ghist[b * 256 + tid], lh[tid]);
}

// ---------------- kernel 2: depth masked reductions ----------------
__global__ __launch_bounds__(256) void mcm_depth_reduce(
    const float* __restrict__ depth, float* __restrict__ dCnt,
    float* __restrict__ dSum, float* __restrict__ dSq)
{
  __shared__ float sm[RT2 * Wd];   // 32 KB staged tile
  __shared__ float red[24];
  const int t = blockIdx.x, b = blockIdx.y, tid = threadIdx.x;

  if (tid < 32) {
    unsigned ldsb = (unsigned)(uintptr_t)(void*)sm;
    tdm_load(ldsb, depth + (size_t)b * HWd + (size_t)t * RT2 * Wd,
             Wd, RT2, 1, Wd, RT2, 1, (unsigned long long)Wd, 0ull);
    __builtin_amdgcn_s_wait_tensorcnt(0);
  }
  __syncthreads();

  float c = 0.f, s = 0.f, q = 0.f;
#pragma unroll 8
  for (int i = 0; i < 32; ++i) {
    float d = sm[tid + i * 256];
    if (d > 0.f) { c += 1.f; s += d; q += d * d; }
  }
  c = wredf(c); s = wredf(s); q = wredf(q);
  int wave = tid >> 5, lane = tid & 31;
  if (lane == 0) { red[wave] = c; red[8 + wave] = s; red[16 + wave] = q; }
  __syncthreads();
  if (tid == 0) {
    float a = 0.f, bb = 0.f, cc = 0.f;
    for (int w = 0; w < 8; ++w) { a += red[w]; bb += red[8 + w]; cc += red[16 + w]; }
    int idx = b * NT2 + t;
    dCnt[idx] = a; dSum[idx] = bb; dSq[idx] = cc;
  }
}

// ---------------- kernel 3: finalize (one wave per batch; WMMA reductions) ----
__global__ __launch_bounds__(32) void mcm_finalize(
    const unsigned* __restrict__ ghist, const float* __restrict__ lapSum,
    const float* __restrict__ lapSq, const float* __restrict__ dCnt,
    const float* __restrict__ dSum, const float* __restrict__ dSq,
    float* __restrict__ out)
{
  const int b = blockIdx.x, lane = threadIdx.x;
  const float N = (float)HWd;

  // entropy terms: 8 bins per lane, then WMMA-tree wave sum
  float et = 0.f;
#pragma unroll
  for (int k = 0; k < 8; ++k) {
    float p = (float)ghist[b * 256 + lane + k * 32] * (1.f / N);
    et += p * log2f(p + EPSF);
  }
  float entropy = -waveSum32(et);

  float s1l = lapSum[b * NT1 + lane] + lapSum[b * NT1 + 32 + lane];
  float s2l = lapSq [b * NT1 + lane] + lapSq [b * NT1 + 32 + lane];
  float S1 = waveSum32(s1l);
  float S2 = waveSum32(s2l);
  float n  = waveSum32(dCnt[b * NT2 + lane]);
  float sd = waveSum32(dSum[b * NT2 + lane]);
  float qd = waveSum32(dSq [b * NT2 + lane]);

  if (lane == 0) {
    float lapVar  = (S2 - S1 * S1 / N) / (N - 1.f);
    float rconf   = 0.5f * (lapVar / (1000.f + EPSF) + 1.f / (entropy + EPSF));
    float density = n / N;
    float mean    = sd / fmaxf(n, 1.f);
    float sq      = qd - 2.f * mean * sd + mean * mean * n;
    float var     = sq / fmaxf(n - 1.f, 1.f);
    float stddev  = sqrtf(fmaxf(var, 0.f));
    float noise   = (n > 0.f) ? stddev : 1.f;
    float dconf   = 0.5f * (density / (10000.f + EPSF) + 1.f / (noise + EPSF));
    float denom   = rconf + dconf + EPSF;
    out[b]      = rconf / denom;   // w_r
    out[Bd + b] = dconf / denom;   // w_d
  }
}

// ---------------- launch ----------------
extern "C" void kernel_launch(void* const* d_in, const int* in_sizes, int n_in,
                              void* d_out, int out_size, void* d_ws, size_t ws_size,
                              hipStream_t stream) {
  (void)in_sizes; (void)n_in; (void)out_size; (void)ws_size;
  const float* rgb   = (const float*)d_in[0];
  const float* depth = (const float*)d_in[1];
  float* out = (float*)d_out;
  float* ws  = (float*)d_ws;

  float*    lapSum = ws;                        // B*64
  float*    lapSq  = ws + Bd * NT1;             // B*64
  unsigned* ghist  = (unsigned*)(lapSq + Bd * NT1);  // B*256
  float*    dCnt   = (float*)(ghist + Bd * 256);     // B*32
  float*    dSum   = dCnt + Bd * NT2;
  float*    dSq    = dSum + Bd * NT2;

  mcm_zero_hist   <<<Bd, 256, 0, stream>>>(ghist);
  mcm_rgb_reduce  <<<dim3(NT1, Bd), 256, 0, stream>>>(rgb, lapSum, lapSq, ghist);
  mcm_depth_reduce<<<dim3(NT2, Bd), 256, 0, stream>>>(depth, dCnt, dSum, dSq);
  mcm_finalize    <<<Bd, 32, 0, stream>>>(ghist, lapSum, lapSq, dCnt, dSum, dSq, out);
}